// KeywordTree_30837865185557
// MI455X (gfx1250) — compile-verified
//
#include <hip/hip_runtime.h>

// ---------------------------------------------------------------------------
// Hierarchical-softmax path loss for MI455X (gfx1250, wave32, WMMA).
//
// h      = outputs[:,0,:]            (64 x 768 f32)
// e      = embedding[path_nodes]     (64 x 48 x 768 f32, gathered)
// dots   = einsum('blh,bh->bl')      via V_WMMA_F32_16X16X4_F32 (B replicated)
// loss   = -mean_b( sum_l log(sigmoid(sign*dot)+eps)*mask / steps_b )
// ---------------------------------------------------------------------------

typedef __attribute__((ext_vector_type(2))) float v2f;
typedef __attribute__((ext_vector_type(8))) float v8f;

#define B_SZ   64
#define S_SZ   512
#define H_SZ   768
#define MAXD   48
#define EPS_F  1e-7f
#define NTILES 3              // 48 levels / 16 per WMMA tile
#define WAVES_PER_BLOCK 4
#define NWAVES (B_SZ * NTILES)   // 192 independent waves

__global__ __launch_bounds__(WAVES_PER_BLOCK * 32)
void ht_dots_kernel(const float* __restrict__ outputs,
                    const float* __restrict__ embedding,
                    const int*   __restrict__ path_nodes,
                    const int*   __restrict__ path_signs,
                    const int*   __restrict__ path_lengths,
                    float*       __restrict__ partial)   // [NWAVES]
{
    const int lane = threadIdx.x & 31;
    const int wave = threadIdx.x >> 5;
    const int w    = blockIdx.x * WAVES_PER_BLOCK + wave;   // 0..191
    const int b    = w / NTILES;
    const int tile = w - b * NTILES;
    const int lb   = tile * 16;                             // level base

    // A-matrix (16x4 f32) layout: lanes 0-15 -> M=lane, K={k0,k0+1};
    //                             lanes 16-31 -> M=lane-16, K={k0+2,k0+3}.
    const int m    = lane & 15;
    const int grp  = lane >> 4;      // 0 or 1
    const int koff = grp * 2;

    const float* __restrict__ hrow = outputs + (size_t)b * (S_SZ * H_SZ); // row s=0
    const int    node              = path_nodes[b * MAXD + lb + m];
    const float* __restrict__ erow = embedding + (size_t)node * H_SZ;

    v8f c = {0.f, 0.f, 0.f, 0.f, 0.f, 0.f, 0.f, 0.f};

    // 768 / 4 = 192 WMMA steps; 2x global_load_b64 per lane per step.
#pragma unroll 8
    for (int k0 = 0; k0 < H_SZ; k0 += 4) {
        v2f a, bb;
        a.x  = erow[k0 + koff];
        a.y  = erow[k0 + koff + 1];
        // B-matrix (4x16 f32): row K striped across lanes; replicate h[k]
        // across all 16 columns so every D column equals the dot product.
        bb.x = hrow[k0 + koff];
        bb.y = hrow[k0 + koff + 1];
        c = __builtin_amdgcn_wmma_f32_16x16x4_f32(
                /*neg_a=*/false, a, /*neg_b=*/false, bb,
                /*c_mod=*/(short)0, c, /*reuse_a=*/false, /*reuse_b=*/false);
    }

    // D layout: VGPR r -> level r (lanes 0-15) / level r+8 (lanes 16-31).
    // Take one copy of each level from column lanes 0..7 of each half-wave.
    float term = 0.f;
    const int idx = lane & 15;
    if (idx < 8) {
        float dot = 0.f;
#pragma unroll
        for (int r = 0; r < 8; ++r) dot = (idx == r) ? c[r] : dot;

        const int lvl  = lb + grp * 8 + idx;
        const int plen = path_lengths[b];
        if (lvl < plen) {
            const float sgn = (float)(2 * path_signs[b * MAXD + lvl] - 1);
            const float x   = sgn * dot;
            const float sig = 1.0f / (1.0f + __expf(-x));   // v_exp_f32
            term = __logf(sig + EPS_F);                      // v_log_f32
        }
    }

    // Wave32 sum reduction.
#pragma unroll
    for (int off = 16; off > 0; off >>= 1)
        term += __shfl_xor(term, off, 32);

    if (lane == 0) partial[w] = term;
}

__global__ __launch_bounds__(B_SZ)
void ht_reduce_kernel(const float* __restrict__ partial,
                      const int*   __restrict__ path_lengths,
                      float*       __restrict__ out)
{
    __shared__ float sm[B_SZ];
    const int b = threadIdx.x;
    const float per = (partial[b * NTILES + 0] +
                       partial[b * NTILES + 1] +
                       partial[b * NTILES + 2]) / (float)path_lengths[b];
    sm[b] = per;
    __syncthreads();
    if (b == 0) {
        float t = 0.f;
#pragma unroll
        for (int i = 0; i < B_SZ; ++i) t += sm[i];
        out[0] = -t / (float)B_SZ;
    }
}

extern "C" void kernel_launch(void* const* d_in, const int* in_sizes, int n_in,
                              void* d_out, int out_size, void* d_ws, size_t ws_size,
                              hipStream_t stream)
{
    (void)in_sizes; (void)n_in; (void)out_size; (void)ws_size;

    const float* outputs      = (const float*)d_in[0];
    const float* embedding    = (const float*)d_in[1];
    const int*   path_nodes   = (const int*)d_in[2];
    const int*   path_signs   = (const int*)d_in[3];
    const int*   path_lengths = (const int*)d_in[4];

    float* out = (float*)d_out;
    float* ws  = (float*)d_ws;     // NWAVES floats of scratch (768 B)

    ht_dots_kernel<<<NWAVES / WAVES_PER_BLOCK, WAVES_PER_BLOCK * 32, 0, stream>>>(
        outputs, embedding, path_nodes, path_signs, path_lengths, ws);

    ht_reduce_kernel<<<1, B_SZ, 0, stream>>>(ws, path_lengths, out);
}